// Graph_Layer_44787918963027
// MI455X (gfx1250) — compile-verified
//
#include <hip/hip_runtime.h>

#define NR   16384
#define DIM  256

typedef __attribute__((ext_vector_type(2))) float v2f;
typedef __attribute__((ext_vector_type(8))) float v8f;

// ws layout (in floats)
#define WS_R      0                    // NR floats: 1/max(||row||,eps)
#define WS_C      16384                // DIM floats: column sums of sf
#define WS_M      16896                // DIM*DIM floats: M = sf^T @ x
#define WS_MPART  (16896 + 65536)      // MCHUNKS * DIM*DIM partials
#define MCHUNKS   8
#define CHUNK_ROWS (NR / MCHUNKS)      // 2048

// ---------- kernel 0: zero the column-sum accumulator ----------
__global__ void k_zero_c(float* ws) {
    ws[WS_C + threadIdx.x] = 0.0f;
}

// ---------- kernel 1: per-row inverse norms + column sums ----------
__global__ void k_norms(const float* __restrict__ sim, float* __restrict__ ws) {
    __shared__ float wred[8];
    __shared__ float rbc;
    const int t    = threadIdx.x;
    const int lane = t & 31;
    const int wid  = t >> 5;
    const int row0 = blockIdx.x * 64;
    float csum = 0.0f;
    for (int rr = 0; rr < 64; ++rr) {
        const int row = row0 + rr;
        const float v = sim[(size_t)row * DIM + t];
        float ss = v * v;
        #pragma unroll
        for (int o = 16; o > 0; o >>= 1) ss += __shfl_xor(ss, o, 32);
        if (lane == 0) wred[wid] = ss;
        __syncthreads();
        if (t == 0) {
            float s = 0.0f;
            #pragma unroll
            for (int w = 0; w < 8; ++w) s += wred[w];
            const float inv = 1.0f / fmaxf(sqrtf(s), 1e-12f);
            ws[WS_R + row] = inv;
            rbc = inv;
        }
        __syncthreads();
        csum += rbc * v;
    }
    atomicAdd(&ws[WS_C + t], csum);
}

// ---------- kernel 2: M partials = sum_i sf_i (x) x_i over a row chunk ----------
__global__ void k_mpart(const float* __restrict__ sim, const float* __restrict__ x,
                        float* __restrict__ ws) {
    __shared__ float sfs[8][64];
    __shared__ float xs [8][64];
    const int t  = threadIdx.x;
    const int tx = t & 15, ty = t >> 4;
    const int c1 = blockIdx.x * 64;
    const int c2 = blockIdx.y * 64;
    const int i0 = blockIdx.z * CHUNK_ROWS;
    const float* r = ws + WS_R;
    float acc[4][4];
    #pragma unroll
    for (int a = 0; a < 4; ++a)
        #pragma unroll
        for (int b = 0; b < 4; ++b) acc[a][b] = 0.0f;

    for (int ib = 0; ib < CHUNK_ROWS; ib += 8) {
        __syncthreads();
        for (int e = t; e < 512; e += 256) {
            const int rr = e >> 6, cc = e & 63;
            const int row = i0 + ib + rr;
            sfs[rr][cc] = r[row] * sim[(size_t)row * DIM + c1 + cc];
            xs [rr][cc] =          x [(size_t)row * DIM + c2 + cc];
        }
        __syncthreads();
        #pragma unroll
        for (int rr = 0; rr < 8; ++rr) {
            float av[4], bv[4];
            #pragma unroll
            for (int a = 0; a < 4; ++a) av[a] = sfs[rr][ty * 4 + a];
            #pragma unroll
            for (int b = 0; b < 4; ++b) bv[b] = xs[rr][tx * 4 + b];
            #pragma unroll
            for (int a = 0; a < 4; ++a)
                #pragma unroll
                for (int b = 0; b < 4; ++b) acc[a][b] += av[a] * bv[b];
        }
    }
    float* mp = ws + WS_MPART + (size_t)blockIdx.z * (DIM * DIM);
    #pragma unroll
    for (int a = 0; a < 4; ++a)
        #pragma unroll
        for (int b = 0; b < 4; ++b)
            mp[(size_t)(c1 + ty * 4 + a) * DIM + (c2 + tx * 4 + b)] = acc[a][b];
}

// ---------- kernel 3: reduce M partials ----------
__global__ void k_mreduce(float* ws) {
    const int i = blockIdx.x * 256 + threadIdx.x;
    float s = 0.0f;
    #pragma unroll
    for (int z = 0; z < MCHUNKS; ++z) s += ws[WS_MPART + (size_t)z * 65536 + i];
    ws[WS_M + i] = s;
}

// ---------- kernel 4: fused WMMA: out = ((sf@M - x) / rowsum) @ weight ----------
// 4 waves / 64 rows per block: each B load (M/weight) feeds 4 WMMAs -> 4x less
// L2 traffic on the B stream. LDS ~130 KB (CDNA5 WGP has 320 KB).
#define LSTR  260   // LDS row stride: 4-bank rotation per row, 16B aligned
#define BROWS 64

__global__ void __launch_bounds__(128) k_fused(
    const float* __restrict__ x, const float* __restrict__ sim,
    const float* __restrict__ weight, const float* __restrict__ ws,
    float* __restrict__ out) {
    __shared__ float sfT[BROWS * LSTR];
    __shared__ float yT [BROWS * LSTR];
    __shared__ float invrs[BROWS];
    const int t    = threadIdx.x;          // 0..127
    const int lane = t & 31;
    const int wv   = t >> 5;               // wave 0..3
    const int row0 = blockIdx.x * BROWS;
    const float* r = ws + WS_R;
    const float* c = ws + WS_C;
    const float* M = ws + WS_M;

    // stage sf tile (64 x 256), pre-scaled by 1/norm
    for (int e = t; e < BROWS * 64; e += 128) {
        const int rr = e >> 6;
        const int c4 = (e & 63) << 2;
        const float4 v = *(const float4*)(sim + (size_t)(row0 + rr) * DIM + c4);
        const float rv = r[row0 + rr];
        float* d = &sfT[rr * LSTR + c4];
        d[0] = rv * v.x; d[1] = rv * v.y; d[2] = rv * v.z; d[3] = rv * v.w;
    }
    __syncthreads();

    // rowsum_i = sf_i . c - 1  ->  invrs  (wave wv handles rows wv*16..+15)
    {
        const int rr = wv * 16 + (lane & 15);
        const int k0 = (lane >> 4) * 128;
        float s = 0.0f;
        for (int k = 0; k < 128; ++k) s += sfT[rr * LSTR + k0 + k] * c[k0 + k];
        s += __shfl_xor(s, 16, 32);
        if (lane < 16) invrs[rr] = 1.0f / (s - 1.0f);
    }
    __syncthreads();

    const int nn   = lane & 15;        // N index in tile / A-matrix row
    const int half = lane >> 4;        // 0: K0/K1 lanes, 1: K2/K3 lanes
    const int kofs = half * 2;

    // ---- stage 1: t = sf @ M ; y = (t - x) * invrs -> yT ----
    for (int jj = 0; jj < 4; ++jj) {
        const int j = jj * 4 + wv;     // this wave's output col-tile
        v8f acc[4];
        #pragma unroll
        for (int rt = 0; rt < 4; ++rt) acc[rt] = (v8f){};
        #pragma unroll 2
        for (int k0 = 0; k0 < DIM; k0 += 4) {
            v2f b;
            const float* bp = &M[(size_t)(k0 + kofs) * DIM + j * 16 + nn];
            b.x = bp[0]; b.y = bp[DIM];
            #pragma unroll
            for (int rt = 0; rt < 4; ++rt) {
                v2f a;
                const float* ap = &sfT[(rt * 16 + nn) * LSTR + k0 + kofs];
                a.x = ap[0]; a.y = ap[1];
                acc[rt] = __builtin_amdgcn_wmma_f32_16x16x4_f32(
                    false, a, false, b, (short)0, acc[rt], false, false);
            }
        }
        #pragma unroll
        for (int rt = 0; rt < 4; ++rt)
            #pragma unroll
            for (int v = 0; v < 8; ++v) {
                const int rr = rt * 16 + v + half * 8;
                const float xv = x[(size_t)(row0 + rr) * DIM + j * 16 + nn];
                yT[rr * LSTR + j * 16 + nn] = (acc[rt][v] - xv) * invrs[rr];
            }
    }
    __syncthreads();

    // ---- stage 2: out = y @ weight ----
    for (int jj = 0; jj < 4; ++jj) {
        const int j = jj * 4 + wv;
        v8f acc[4];
        #pragma unroll
        for (int rt = 0; rt < 4; ++rt) acc[rt] = (v8f){};
        #pragma unroll 2
        for (int k0 = 0; k0 < DIM; k0 += 4) {
            v2f b;
            const float* bp = &weight[(size_t)(k0 + kofs) * DIM + j * 16 + nn];
            b.x = bp[0]; b.y = bp[DIM];
            #pragma unroll
            for (int rt = 0; rt < 4; ++rt) {
                v2f a;
                const float* ap = &yT[(rt * 16 + nn) * LSTR + k0 + kofs];
                a.x = ap[0]; a.y = ap[1];
                acc[rt] = __builtin_amdgcn_wmma_f32_16x16x4_f32(
                    false, a, false, b, (short)0, acc[rt], false, false);
            }
        }
        #pragma unroll
        for (int rt = 0; rt < 4; ++rt)
            #pragma unroll
            for (int v = 0; v < 8; ++v) {
                const int rr = rt * 16 + v + half * 8;
                out[(size_t)(row0 + rr) * DIM + j * 16 + nn] = acc[rt][v];
            }
    }
}

extern "C" void kernel_launch(void* const* d_in, const int* in_sizes, int n_in,
                              void* d_out, int out_size, void* d_ws, size_t ws_size,
                              hipStream_t stream) {
    const float* x      = (const float*)d_in[0];
    const float* sim    = (const float*)d_in[1];
    // d_in[2] is k (unused by the reference math)
    const float* weight = (const float*)d_in[3];
    float* out = (float*)d_out;
    float* ws  = (float*)d_ws;   // uses ~2.43 MB

    k_zero_c<<<1, DIM, 0, stream>>>(ws);
    k_norms<<<NR / 64, 256, 0, stream>>>(sim, ws);
    dim3 g3(4, 4, MCHUNKS);
    k_mpart<<<g3, 256, 0, stream>>>(sim, x, ws);
    k_mreduce<<<(DIM * DIM) / 256, 256, 0, stream>>>(ws);
    k_fused<<<NR / BROWS, 128, 0, stream>>>(x, sim, weight, ws, out);
}